// GAT_21242908246157
// MI455X (gfx1250) — compile-verified
//
#include <hip/hip_runtime.h>
#include <math.h>

typedef float v2f __attribute__((ext_vector_type(2)));
typedef float v4f __attribute__((ext_vector_type(4)));
typedef float v8f __attribute__((ext_vector_type(8)));

#define IN_CH   128
#define HID     32
#define HEADS   4
#define OUT_CH  64
#define NEG_SLOPE 0.2f
#define NEG_INF  -3.0e38f

// Native f32 atomics (GLOBAL_ATOMIC_ADD_F32 / GLOBAL_ATOMIC_MAX_NUM_F32),
// avoiding the CAS-loop fallback of the generic overloads.
__device__ __forceinline__ float atomAddF(float* p, float v) {
#if defined(__HIP_PLATFORM_AMD__) || defined(__AMDGCN__)
  return unsafeAtomicAdd(p, v);
#else
  return atomicAdd(p, v);
#endif
}
__device__ __forceinline__ float atomMaxF(float* p, float v) {
#if defined(__HIP_PLATFORM_AMD__) || defined(__AMDGCN__)
  return unsafeAtomicMax(p, v);
#else
  return atomicMax(p, v);
#endif
}

// ---------------------------------------------------------------------------
// WMMA fp32 GEMM: C[M,NC] = A[M,K] @ B[K,NC]  (row-major, M%16==0, K%4==0,
// NC%64==0). One wave computes a 16x64 strip: one A fragment feeds 4
// independent V_WMMA_F32_16X16X4_F32 accumulators (A reuse x4, and 4
// independent WMMAs per k-step to cover the XDL pipeline).
// A 16x4 layout: lanes 0-15 -> M=lane, VGPR0=K0,VGPR1=K1; lanes 16-31 -> K2,K3.
// B 4x16 layout: lanes 0-15 -> N=lane, VGPR0=K0,VGPR1=K1; lanes 16-31 -> K2,K3.
// D 16x16 f32:   VGPR r -> row r (lanes 0-15) / row r+8 (lanes 16-31), col=lane&15.
// ---------------------------------------------------------------------------
__global__ void k_gemm_wmma(const float* __restrict__ A, const float* __restrict__ B,
                            float* __restrict__ C, int M, int K, int NC) {
  const int lane = threadIdx.x & 31;
  const int wid  = blockIdx.x * (blockDim.x >> 5) + (threadIdx.x >> 5);
  const int ngrp = NC >> 6;                      // 64-column groups
  const int strips = (M >> 4) * ngrp;
  if (wid >= strips) return;
  const int tm = wid / ngrp, tg = wid - tm * ngrp;
  const int m0 = tm << 4, n0 = tg << 6;
  const int arow = m0 + (lane & 15);
  const int ak   = (lane >> 4) << 1;             // 0 or 2
  const int bn   = n0 + (lane & 15);
  const float* __restrict__ Arow = A + (size_t)arow * K;
  v8f acc0 = {}, acc1 = {}, acc2 = {}, acc3 = {};
  for (int k = 0; k < K; k += 4) {
    v2f a = *(const v2f*)(Arow + k + ak);        // 8B aligned: k+ak even
    const float* __restrict__ B0 = B + (size_t)(k + ak) * NC + bn;
    const float* __restrict__ B1 = B0 + NC;
    v2f b0 = {B0[0],  B1[0]},  b1 = {B0[16], B1[16]};
    v2f b2 = {B0[32], B1[32]}, b3 = {B0[48], B1[48]};
    acc0 = __builtin_amdgcn_wmma_f32_16x16x4_f32(false, a, false, b0, (short)0, acc0, false, false);
    acc1 = __builtin_amdgcn_wmma_f32_16x16x4_f32(false, a, false, b1, (short)0, acc1, false, false);
    acc2 = __builtin_amdgcn_wmma_f32_16x16x4_f32(false, a, false, b2, (short)0, acc2, false, false);
    acc3 = __builtin_amdgcn_wmma_f32_16x16x4_f32(false, a, false, b3, (short)0, acc3, false, false);
  }
  const int rbase = m0 + ((lane >> 4) << 3);
  float* __restrict__ Cp = C + (size_t)rbase * NC + bn;
#pragma unroll
  for (int r = 0; r < 8; ++r) {
    Cp[(size_t)r * NC +  0] = acc0[r];
    Cp[(size_t)r * NC + 16] = acc1[r];
    Cp[(size_t)r * NC + 32] = acc2[r];
    Cp[(size_t)r * NC + 48] = acc3[r];
  }
}

// Fused A-fetch for layer-2 GEMM: h = relu(concat(out_ppi,out_go)+b1), never
// materialized. k in [0,256).
__device__ __forceinline__ float fetch_h(const float* __restrict__ op,
                                         const float* __restrict__ og,
                                         const float* __restrict__ b1,
                                         int row, int k) {
  float v = (k < IN_CH) ? (op[(size_t)row * IN_CH + k] + b1[k])
                        : (og[(size_t)row * IN_CH + (k - IN_CH)] + b1[k - IN_CH]);
  return v > 0.f ? v : 0.f;
}

__global__ void k_gemm2_wmma(const float* __restrict__ out_ppi,
                             const float* __restrict__ out_go,
                             const float* __restrict__ b1,
                             const float* __restrict__ B,
                             float* __restrict__ C, int M, int K, int NC) {
  const int lane = threadIdx.x & 31;
  const int wid  = blockIdx.x * (blockDim.x >> 5) + (threadIdx.x >> 5);
  const int ngrp = NC >> 6;
  const int strips = (M >> 4) * ngrp;
  if (wid >= strips) return;
  const int tm = wid / ngrp, tg = wid - tm * ngrp;
  const int m0 = tm << 4, n0 = tg << 6;
  const int arow = m0 + (lane & 15);
  const int ak   = (lane >> 4) << 1;
  const int bn   = n0 + (lane & 15);
  v8f acc0 = {}, acc1 = {}, acc2 = {}, acc3 = {};
  for (int k = 0; k < K; k += 4) {
    v2f a;
    a.x = fetch_h(out_ppi, out_go, b1, arow, k + ak);
    a.y = fetch_h(out_ppi, out_go, b1, arow, k + ak + 1);
    const float* __restrict__ B0 = B + (size_t)(k + ak) * NC + bn;
    const float* __restrict__ B1 = B0 + NC;
    v2f b0 = {B0[0],  B1[0]},  b1v = {B0[16], B1[16]};
    v2f b2 = {B0[32], B1[32]}, b3  = {B0[48], B1[48]};
    acc0 = __builtin_amdgcn_wmma_f32_16x16x4_f32(false, a, false, b0,  (short)0, acc0, false, false);
    acc1 = __builtin_amdgcn_wmma_f32_16x16x4_f32(false, a, false, b1v, (short)0, acc1, false, false);
    acc2 = __builtin_amdgcn_wmma_f32_16x16x4_f32(false, a, false, b2,  (short)0, acc2, false, false);
    acc3 = __builtin_amdgcn_wmma_f32_16x16x4_f32(false, a, false, b3,  (short)0, acc3, false, false);
  }
  const int rbase = m0 + ((lane >> 4) << 3);
  float* __restrict__ Cp = C + (size_t)rbase * NC + bn;
#pragma unroll
  for (int r = 0; r < 8; ++r) {
    Cp[(size_t)r * NC +  0] = acc0[r];
    Cp[(size_t)r * NC + 16] = acc1[r];
    Cp[(size_t)r * NC + 32] = acc2[r];
    Cp[(size_t)r * NC + 48] = acc3[r];
  }
}

// a_src[n,h] = sum_c xp[n,h,c]*att_src[h,c];  a_dst likewise. C%4==0, rows
// 16B-aligned -> float4 loads.
__global__ void k_att(const float* __restrict__ xp,
                      const float* __restrict__ att_s, const float* __restrict__ att_d,
                      float* __restrict__ asrc, float* __restrict__ adst,
                      int Nn, int H, int C) {
  int t = blockIdx.x * blockDim.x + threadIdx.x;
  if (t >= Nn * H) return;
  int n = t / H, h = t - n * H;
  const float* __restrict__ row = xp + (size_t)n * H * C + (size_t)h * C;
  const float* __restrict__ as  = att_s + (size_t)h * C;
  const float* __restrict__ ad  = att_d + (size_t)h * C;
  float ss = 0.f, sd = 0.f;
  for (int c = 0; c < C; c += 4) {
    v4f v = *(const v4f*)(row + c);
    v4f s = *(const v4f*)(as + c);
    v4f d = *(const v4f*)(ad + c);
    ss += v.x * s.x + v.y * s.y + v.z * s.z + v.w * s.w;
    sd += v.x * d.x + v.y * d.y + v.z * d.z + v.w * d.w;
  }
  asrc[t] = ss; adst[t] = sd;
}

__global__ void k_fill(float* __restrict__ p, float v, size_t n) {
  size_t i = (size_t)blockIdx.x * blockDim.x + threadIdx.x;
  size_t stride = (size_t)gridDim.x * blockDim.x;
  for (; i < n; i += stride) p[i] = v;
}

// Seed output accumulator with the bias (aggregation then atomically adds).
__global__ void k_init_bias(float* __restrict__ out, const float* __restrict__ b,
                            int Nn, int C) {
  size_t i = (size_t)blockIdx.x * blockDim.x + threadIdx.x;
  size_t n = (size_t)Nn * C;
  size_t stride = (size_t)gridDim.x * blockDim.x;
  for (; i < n; i += stride) out[i] = b[i % C];
}

__device__ __forceinline__ void edge_sd(const int* __restrict__ srcI,
                                        const int* __restrict__ dstI,
                                        int e, int E_, int& s, int& d) {
  if (e < E_) { s = srcI[e]; d = dstI[e]; }
  else        { s = e - E_;  d = s; }        // self loops
}

// Pass 1: e = leaky_relu(a_src[src]+a_dst[dst]); store raw e; running max per dst.
__global__ void k_edge_logits(const int* __restrict__ srcI, const int* __restrict__ dstI,
                              int E_, int Etot,
                              const float* __restrict__ asrc, const float* __restrict__ adst,
                              float* __restrict__ ebuf, float* __restrict__ mmax, int H) {
  int e = blockIdx.x * blockDim.x + threadIdx.x;
  if (e >= Etot) return;
  int s, d; edge_sd(srcI, dstI, e, E_, s, d);
  for (int h = 0; h < H; ++h) {
    float v = asrc[(size_t)s * H + h] + adst[(size_t)d * H + h];
    v = v > 0.f ? v : NEG_SLOPE * v;
    ebuf[(size_t)e * H + h] = v;
    atomMaxF(&mmax[(size_t)d * H + h], v);   // GLOBAL_ATOMIC_MAX_NUM_F32
  }
}

// Pass 2: ex = exp(e - m[dst]); store in place; denom[dst] += ex.
__global__ void k_edge_exp(const int* __restrict__ srcI, const int* __restrict__ dstI,
                           int E_, int Etot,
                           float* __restrict__ ebuf, const float* __restrict__ mmax,
                           float* __restrict__ denom, int H) {
  int e = blockIdx.x * blockDim.x + threadIdx.x;
  if (e >= Etot) return;
  int s, d; edge_sd(srcI, dstI, e, E_, s, d);
  for (int h = 0; h < H; ++h) {
    float ex = __expf(ebuf[(size_t)e * H + h] - mmax[(size_t)d * H + h]);
    ebuf[(size_t)e * H + h] = ex;
    atomAddF(&denom[(size_t)d * H + h], ex);
  }
}

// Pass 3: wave per edge; lanes stripe channels; out[dst] += xp[src]*alpha.
__global__ void k_edge_aggr(const int* __restrict__ srcI, const int* __restrict__ dstI,
                            int E_, int Etot,
                            const float* __restrict__ xp, const float* __restrict__ ebuf,
                            const float* __restrict__ denom, float* __restrict__ out,
                            int H, int C) {
  int lane = threadIdx.x & 31;
  int w = blockIdx.x * (blockDim.x >> 5) + (threadIdx.x >> 5);
  if (w >= Etot) return;
  int s, d; edge_sd(srcI, dstI, w, E_, s, d);
  const int HC = H * C;
  for (int h = 0; h < H; ++h) {
    float alpha = ebuf[(size_t)w * H + h] / denom[(size_t)d * H + h];
    for (int c = lane; c < C; c += 32) {
      int idx = h * C + c;
      atomAddF(&out[(size_t)d * HC + idx], xp[(size_t)s * HC + idx] * alpha);
    }
  }
}

// ---------------------------------------------------------------------------
extern "C" void kernel_launch(void* const* d_in, const int* in_sizes, int n_in,
                              void* d_out, int out_size, void* d_ws, size_t ws_size,
                              hipStream_t stream) {
  const float* x_ppi    = (const float*)d_in[0];
  const float* x_go     = (const float*)d_in[1];
  const int*   ei       = (const int*)d_in[2];
  const float* W1       = (const float*)d_in[3];
  const float* att_src1 = (const float*)d_in[4];
  const float* att_dst1 = (const float*)d_in[5];
  const float* b1       = (const float*)d_in[6];
  const float* W2       = (const float*)d_in[7];
  const float* att_src2 = (const float*)d_in[8];
  const float* att_dst2 = (const float*)d_in[9];
  const float* b2       = (const float*)d_in[10];
  float* out = (float*)d_out;

  const int Nn   = in_sizes[0] / IN_CH;     // 100000
  const int E_   = in_sizes[2] / 2;         // 1600000
  const int Etot = E_ + Nn;                 // + self loops
  const int* srcI = ei;
  const int* dstI = ei + E_;

  // ---- workspace carve-up (floats), ~262 MB total ----
  float* ws = (float*)d_ws;
  size_t off = 0;
  float* xp_ppi  = ws + off; off += (size_t)Nn * 128;
  float* xp_go   = ws + off; off += (size_t)Nn * 128;
  float* out_ppi = ws + off; off += (size_t)Nn * 128;
  float* out_go  = ws + off; off += (size_t)Nn * 128;
  float* xp2     = ws + off; off += (size_t)Nn * 64;
  float* asrc    = ws + off; off += (size_t)Nn * HEADS;
  float* adst    = ws + off; off += (size_t)Nn * HEADS;
  float* mmax    = ws + off; off += (size_t)Nn * HEADS;
  float* denom   = ws + off; off += (size_t)Nn * HEADS;
  float* ebuf    = ws + off; off += (size_t)Etot * HEADS;
  (void)ws_size;

  const int BLK = 256;
  const int waves_per_blk = BLK / 32;
  // GEMM1: (Nn/16) * (128/64) 16x64-strips
  int g1_waves  = (Nn / 16) * (128 / 64);
  int g1_blocks = (g1_waves + waves_per_blk - 1) / waves_per_blk;
  // GEMM2: (Nn/16) * (64/64) strips
  int g2_waves  = (Nn / 16) * (64 / 64);
  int g2_blocks = (g2_waves + waves_per_blk - 1) / waves_per_blk;

  int edge_blocks = (Etot + BLK - 1) / BLK;
  int aggr_blocks = (Etot + waves_per_blk - 1) / waves_per_blk;
  int att1_blocks = (Nn * HEADS + BLK - 1) / BLK;
  int att2_blocks = (Nn + BLK - 1) / BLK;
  int fill_blocks = 2048;

  // ================= Layer 1 (shared W1/attn) on both feature sets =========
  const float* xs[2] = { x_ppi, x_go };
  float* xps[2]      = { xp_ppi, xp_go };
  float* outs1[2]    = { out_ppi, out_go };
  for (int i = 0; i < 2; ++i) {
    k_gemm_wmma<<<g1_blocks, BLK, 0, stream>>>(xs[i], W1, xps[i], Nn, IN_CH, HEADS * HID);
    k_att<<<att1_blocks, BLK, 0, stream>>>(xps[i], att_src1, att_dst1, asrc, adst,
                                           Nn, HEADS, HID);
    k_fill<<<fill_blocks, BLK, 0, stream>>>(mmax,  NEG_INF, (size_t)Nn * HEADS);
    k_fill<<<fill_blocks, BLK, 0, stream>>>(denom, 0.f,     (size_t)Nn * HEADS);
    k_fill<<<fill_blocks, BLK, 0, stream>>>(outs1[i], 0.f,  (size_t)Nn * 128);
    k_edge_logits<<<edge_blocks, BLK, 0, stream>>>(srcI, dstI, E_, Etot, asrc, adst,
                                                   ebuf, mmax, HEADS);
    k_edge_exp<<<edge_blocks, BLK, 0, stream>>>(srcI, dstI, E_, Etot, ebuf, mmax,
                                                denom, HEADS);
    k_edge_aggr<<<aggr_blocks, BLK, 0, stream>>>(srcI, dstI, E_, Etot, xps[i], ebuf,
                                                 denom, outs1[i], HEADS, HID);
  }

  // ================= Layer 2: h = relu(concat(out_ppi,out_go)+b1) fused ====
  k_gemm2_wmma<<<g2_blocks, BLK, 0, stream>>>(out_ppi, out_go, b1, W2, xp2,
                                              Nn, 2 * HEADS * HID, OUT_CH);
  k_att<<<att2_blocks, BLK, 0, stream>>>(xp2, att_src2, att_dst2, asrc, adst,
                                         Nn, 1, OUT_CH);
  k_fill<<<fill_blocks, BLK, 0, stream>>>(mmax,  NEG_INF, (size_t)Nn);
  k_fill<<<fill_blocks, BLK, 0, stream>>>(denom, 0.f,     (size_t)Nn);
  k_init_bias<<<fill_blocks, BLK, 0, stream>>>(out, b2, Nn, OUT_CH);
  k_edge_logits<<<edge_blocks, BLK, 0, stream>>>(srcI, dstI, E_, Etot, asrc, adst,
                                                 ebuf, mmax, 1);
  k_edge_exp<<<edge_blocks, BLK, 0, stream>>>(srcI, dstI, E_, Etot, ebuf, mmax,
                                              denom, 1);
  k_edge_aggr<<<aggr_blocks, BLK, 0, stream>>>(srcI, dstI, E_, Etot, xp2, ebuf,
                                               denom, out, 1, OUT_CH);
}